// GAT_22574348108053
// MI455X (gfx1250) — compile-verified
//
#include <hip/hip_runtime.h>
#include <hip/hip_bf16.h>

// ---------------- problem constants (from reference) ----------------
#define NN      100000      // nodes
#define NE      1600000     // edges
#define NFEAT   256
#define HEADS   8
#define NHID    8
#define H1DIM   64          // HEADS*NHID
#define NCLASS  40
#define NCPAD   48          // 3 col tiles of 16
#define NEG_SLOPE 0.2f
#define ENC_NEG_INF 0x007FFFFFu   // ordered-uint encoding of -inf

typedef float v2f __attribute__((ext_vector_type(2)));
typedef float v8f __attribute__((ext_vector_type(8)));

__device__ __forceinline__ float lrelu(float v) { return v > 0.f ? v : NEG_SLOPE * v; }

// monotone float <-> uint mapping so unsigned atomicMax == float max
__device__ __forceinline__ unsigned enc_ord(float f) {
  unsigned u = __float_as_uint(f);
  return (u & 0x80000000u) ? ~u : (u | 0x80000000u);
}
__device__ __forceinline__ float dec_ord(unsigned u) {
  return __uint_as_float((u & 0x80000000u) ? (u & 0x7FFFFFFFu) : ~u);
}

// ---------------- GEMM1: h1[NN,64] = x[NN,256] @ W1[256,64] (f32 WMMA) -------
// One wave -> one 16x64 row strip (4 accumulators). A fragments loaded once
// per k-step, reused across 4 col tiles. Strips = NN/16 = 6250.
__global__ void k_gemm1(const float* __restrict__ x, const float* __restrict__ W1,
                        float* __restrict__ h1) {
  const int wave = threadIdx.x >> 5;
  const int lane = threadIdx.x & 31;
  const int strip = blockIdx.x * 8 + wave;
  if (strip >= NN / 16) return;             // wave-uniform
  const int row0 = strip * 16;
  const int m  = lane & 15;                 // row within tile (A) / col (B,D)
  const int kh = lane >> 4;                 // k-half select

  v8f acc0 = {}, acc1 = {}, acc2 = {}, acc3 = {};
  const float* xrow = x  + (row0 + m) * NFEAT + 2 * kh;
  const float* wcol = W1 + (2 * kh) * H1DIM + m;      // + tile*16 per col tile
  #pragma unroll 4
  for (int k0 = 0; k0 < NFEAT; k0 += 4) {
    v2f a = *(const v2f*)(xrow + k0);                 // A[m][k0+2kh .. +1]
    const float* wk = wcol + k0 * H1DIM;
    v2f b0, b1, b2, b3;
    b0.x = wk[ 0];          b0.y = wk[H1DIM +  0];
    b1.x = wk[16];          b1.y = wk[H1DIM + 16];
    b2.x = wk[32];          b2.y = wk[H1DIM + 32];
    b3.x = wk[48];          b3.y = wk[H1DIM + 48];
    acc0 = __builtin_amdgcn_wmma_f32_16x16x4_f32(false, a, false, b0, (short)0, acc0, false, false);
    acc1 = __builtin_amdgcn_wmma_f32_16x16x4_f32(false, a, false, b1, (short)0, acc1, false, false);
    acc2 = __builtin_amdgcn_wmma_f32_16x16x4_f32(false, a, false, b2, (short)0, acc2, false, false);
    acc3 = __builtin_amdgcn_wmma_f32_16x16x4_f32(false, a, false, b3, (short)0, acc3, false, false);
  }
  float* o = h1 + (row0 + 8 * kh) * H1DIM + m;        // D: VGPR r -> row r+8*kh
  #pragma unroll
  for (int r = 0; r < 8; ++r) {
    o[r * H1DIM +  0] = acc0[r];
    o[r * H1DIM + 16] = acc1[r];
    o[r * H1DIM + 32] = acc2[r];
    o[r * H1DIM + 48] = acc3[r];
  }
}

// ---------------- per-node attention logits, layer 1 ------------------------
__global__ void k_alpha1(const float* __restrict__ h1, const float* __restrict__ a_src,
                         const float* __restrict__ a_dst,
                         float* __restrict__ as1, float* __restrict__ ad1) {
  int tid = blockIdx.x * blockDim.x + threadIdx.x;    // NN*HEADS threads
  if (tid >= NN * HEADS) return;
  int n = tid >> 3, h = tid & 7;
  const float* hp = h1 + n * H1DIM + h * NHID;
  float s = 0.f, d = 0.f;
  #pragma unroll
  for (int c = 0; c < NHID; ++c) {
    s += hp[c] * a_src[h * NHID + c];
    d += hp[c] * a_dst[h * NHID + c];
  }
  as1[tid] = s;
  ad1[tid] = d;
}

// ---------------- init layer-1 accumulators ---------------------------------
__global__ void k_init1(float* __restrict__ acc1, unsigned* __restrict__ m1,
                        float* __restrict__ den1) {
  int tid = blockIdx.x * blockDim.x + threadIdx.x;    // NN*64 threads
  if (tid >= NN * H1DIM) return;
  acc1[tid] = 0.f;
  if (tid < NN * HEADS) { m1[tid] = ENC_NEG_INF; den1[tid] = 0.f; }
}

// ---------------- edge pass 1: segment max ----------------------------------
__global__ void k_edge_max1(const int* __restrict__ ei, const float* __restrict__ as1,
                            const float* __restrict__ ad1, unsigned* __restrict__ m1) {
  int e = blockIdx.x * blockDim.x + threadIdx.x;
  if (e >= NE) return;
  int s = ei[e], d = ei[NE + e];
  #pragma unroll
  for (int h = 0; h < HEADS; ++h) {
    float ev = lrelu(as1[s * HEADS + h] + ad1[d * HEADS + h]);
    atomicMax(&m1[d * HEADS + h], enc_ord(ev));
  }
}

// ---------------- edge pass 2: softmax denominator --------------------------
__global__ void k_edge_den1(const int* __restrict__ ei, const float* __restrict__ as1,
                            const float* __restrict__ ad1, const unsigned* __restrict__ m1,
                            float* __restrict__ den1) {
  int e = blockIdx.x * blockDim.x + threadIdx.x;
  if (e >= NE) return;
  int s = ei[e], d = ei[NE + e];
  #pragma unroll
  for (int h = 0; h < HEADS; ++h) {
    float ev = lrelu(as1[s * HEADS + h] + ad1[d * HEADS + h]);
    atomicAdd(&den1[d * HEADS + h], __expf(ev - dec_ord(m1[d * HEADS + h])));
  }
}

// ---------------- edge pass 3: weighted scatter-sum (one thread per edge,head)
__global__ void k_edge_scat1(const int* __restrict__ ei, const float* __restrict__ as1,
                             const float* __restrict__ ad1, const unsigned* __restrict__ m1,
                             const float* __restrict__ den1, const float* __restrict__ h1,
                             float* __restrict__ acc1) {
  int tid = blockIdx.x * blockDim.x + threadIdx.x;    // NE*HEADS threads
  if (tid >= NE * HEADS) return;
  int e = tid >> 3, h = tid & 7;
  int s = ei[e], d = ei[NE + e];
  float ev = lrelu(as1[s * HEADS + h] + ad1[d * HEADS + h]);
  float ex = __expf(ev - dec_ord(m1[d * HEADS + h]));
  float alpha = ex / (den1[d * HEADS + h] + 1e-16f);
  const float4* hp = (const float4*)(h1 + s * H1DIM + h * NHID);
  float4 v0 = hp[0], v1 = hp[1];
  float* o = acc1 + d * H1DIM + h * NHID;
  atomicAdd(o + 0, v0.x * alpha); atomicAdd(o + 1, v0.y * alpha);
  atomicAdd(o + 2, v0.z * alpha); atomicAdd(o + 3, v0.w * alpha);
  atomicAdd(o + 4, v1.x * alpha); atomicAdd(o + 5, v1.y * alpha);
  atomicAdd(o + 6, v1.z * alpha); atomicAdd(o + 7, v1.w * alpha);
}

// ---------------- finalize layer 1: +b1 then ELU (in place -> x2) -----------
__global__ void k_elu1(float* __restrict__ acc1, const float* __restrict__ b1) {
  int tid = blockIdx.x * blockDim.x + threadIdx.x;    // NN*64 threads
  if (tid >= NN * H1DIM) return;
  float v = acc1[tid] + b1[tid & (H1DIM - 1)];
  acc1[tid] = v > 0.f ? v : expm1f(v);
}

// ---------------- GEMM2: h2[NN,40] = x2[NN,64] @ W2[64,40] (padded to 48) ---
// One wave -> 16x48 row strip, 3 accumulators, K = 64 -> 16 k-steps.
__global__ void k_gemm2(const float* __restrict__ x2, const float* __restrict__ W2,
                        float* __restrict__ h2) {
  const int wave = threadIdx.x >> 5;
  const int lane = threadIdx.x & 31;
  const int strip = blockIdx.x * 8 + wave;
  if (strip >= NN / 16) return;             // wave-uniform
  const int row0 = strip * 16;
  const int m  = lane & 15;
  const int kh = lane >> 4;

  v8f acc0 = {}, acc1 = {}, acc2 = {};
  const float* xrow = x2 + (row0 + m) * H1DIM + 2 * kh;
  #pragma unroll 4
  for (int k0 = 0; k0 < H1DIM; k0 += 4) {
    v2f a = *(const v2f*)(xrow + k0);
    int kk = k0 + 2 * kh;
    const float* wk = W2 + kk * NCLASS + m;
    v2f b0, b1, b2;
    b0.x = wk[0];                           b0.y = wk[NCLASS +  0];
    b1.x = wk[16];                          b1.y = wk[NCLASS + 16];
    b2.x = (m + 32 < NCLASS) ? wk[32]           : 0.f;
    b2.y = (m + 32 < NCLASS) ? wk[NCLASS + 32]  : 0.f;
    acc0 = __builtin_amdgcn_wmma_f32_16x16x4_f32(false, a, false, b0, (short)0, acc0, false, false);
    acc1 = __builtin_amdgcn_wmma_f32_16x16x4_f32(false, a, false, b1, (short)0, acc1, false, false);
    acc2 = __builtin_amdgcn_wmma_f32_16x16x4_f32(false, a, false, b2, (short)0, acc2, false, false);
  }
  float* o = h2 + (row0 + 8 * kh) * NCLASS + m;
  #pragma unroll
  for (int r = 0; r < 8; ++r) {
    o[r * NCLASS +  0] = acc0[r];
    o[r * NCLASS + 16] = acc1[r];
    if (m + 32 < NCLASS) o[r * NCLASS + 32] = acc2[r];
  }
}

// ---------------- per-node logits + init, layer 2 ---------------------------
__global__ void k_alpha2(const float* __restrict__ h2, const float* __restrict__ a_src,
                         const float* __restrict__ a_dst, float* __restrict__ as2,
                         float* __restrict__ ad2, unsigned* __restrict__ m2,
                         float* __restrict__ den2) {
  int n = blockIdx.x * blockDim.x + threadIdx.x;
  if (n >= NN) return;
  const float* hp = h2 + n * NCLASS;
  float s = 0.f, d = 0.f;
  #pragma unroll 8
  for (int c = 0; c < NCLASS; ++c) { s += hp[c] * a_src[c]; d += hp[c] * a_dst[c]; }
  as2[n] = s; ad2[n] = d; m2[n] = ENC_NEG_INF; den2[n] = 0.f;
}

// ---------------- init output with bias b2 ----------------------------------
__global__ void k_outinit(float* __restrict__ out, const float* __restrict__ b2) {
  int tid = blockIdx.x * blockDim.x + threadIdx.x;    // NN*NCLASS threads
  if (tid >= NN * NCLASS) return;
  out[tid] = b2[tid % NCLASS];
}

__global__ void k_edge_max2(const int* __restrict__ ei, const float* __restrict__ as2,
                            const float* __restrict__ ad2, unsigned* __restrict__ m2) {
  int e = blockIdx.x * blockDim.x + threadIdx.x;
  if (e >= NE) return;
  int s = ei[e], d = ei[NE + e];
  atomicMax(&m2[d], enc_ord(lrelu(as2[s] + ad2[d])));
}

__global__ void k_edge_den2(const int* __restrict__ ei, const float* __restrict__ as2,
                            const float* __restrict__ ad2, const unsigned* __restrict__ m2,
                            float* __restrict__ den2) {
  int e = blockIdx.x * blockDim.x + threadIdx.x;
  if (e >= NE) return;
  int s = ei[e], d = ei[NE + e];
  atomicAdd(&den2[d], __expf(lrelu(as2[s] + ad2[d]) - dec_ord(m2[d])));
}

// one thread per (edge, 8-channel chunk): NE*5 threads
__global__ void k_edge_scat2(const int* __restrict__ ei, const float* __restrict__ as2,
                             const float* __restrict__ ad2, const unsigned* __restrict__ m2,
                             const float* __restrict__ den2, const float* __restrict__ h2,
                             float* __restrict__ out) {
  int tid = blockIdx.x * blockDim.x + threadIdx.x;
  if (tid >= NE * 5) return;
  int e = tid / 5, c0 = (tid % 5) * 8;
  int s = ei[e], d = ei[NE + e];
  float ev = lrelu(as2[s] + ad2[d]);
  float ex = __expf(ev - dec_ord(m2[d]));
  float alpha = ex / (den2[d] + 1e-16f);
  const float4* hp = (const float4*)(h2 + s * NCLASS + c0);
  float4 v0 = hp[0], v1 = hp[1];
  float* o = out + d * NCLASS + c0;
  atomicAdd(o + 0, v0.x * alpha); atomicAdd(o + 1, v0.y * alpha);
  atomicAdd(o + 2, v0.z * alpha); atomicAdd(o + 3, v0.w * alpha);
  atomicAdd(o + 4, v1.x * alpha); atomicAdd(o + 5, v1.y * alpha);
  atomicAdd(o + 6, v1.z * alpha); atomicAdd(o + 7, v1.w * alpha);
}

// ---------------------------------------------------------------------------
extern "C" void kernel_launch(void* const* d_in, const int* in_sizes, int n_in,
                              void* d_out, int out_size, void* d_ws, size_t ws_size,
                              hipStream_t stream) {
  const float* x      = (const float*)d_in[0];
  const int*   ei     = (const int*)  d_in[1];   // edge_index [2, NE]
  const float* W1     = (const float*)d_in[2];
  const float* a1_src = (const float*)d_in[3];
  const float* a1_dst = (const float*)d_in[4];
  const float* b1     = (const float*)d_in[5];
  const float* W2     = (const float*)d_in[6];
  const float* a2_src = (const float*)d_in[7];
  const float* a2_dst = (const float*)d_in[8];
  const float* b2     = (const float*)d_in[9];
  float* out = (float*)d_out;

  // workspace carve-up (204 floats/node ~= 82 MB)
  float*    h1   = (float*)d_ws;            // NN*64
  float*    as1  = h1  + NN * H1DIM;        // NN*8
  float*    ad1  = as1 + NN * HEADS;        // NN*8
  unsigned* m1   = (unsigned*)(ad1 + NN * HEADS);       // NN*8
  float*    den1 = (float*)(m1 + NN * HEADS);           // NN*8
  float*    acc1 = den1 + NN * HEADS;       // NN*64 (becomes x2 after ELU)
  float*    h2   = acc1 + NN * H1DIM;       // NN*40
  float*    as2  = h2  + NN * NCLASS;       // NN
  float*    ad2  = as2 + NN;                // NN
  unsigned* m2   = (unsigned*)(ad2 + NN);   // NN
  float*    den2 = (float*)(m2 + NN);       // NN

  const int B = 256;
  #define GRID(n) (((n) + B - 1) / B)
  const int strips = NN / 16;               // 6250
  const int strip_blocks = (strips + 7) / 8;

  // ---- layer 1 ----
  k_gemm1<<<strip_blocks, B, 0, stream>>>(x, W1, h1);
  k_alpha1<<<GRID(NN * HEADS), B, 0, stream>>>(h1, a1_src, a1_dst, as1, ad1);
  k_init1<<<GRID(NN * H1DIM), B, 0, stream>>>(acc1, m1, den1);
  k_edge_max1<<<GRID(NE), B, 0, stream>>>(ei, as1, ad1, m1);
  k_edge_den1<<<GRID(NE), B, 0, stream>>>(ei, as1, ad1, m1, den1);
  k_edge_scat1<<<GRID(NE * HEADS), B, 0, stream>>>(ei, as1, ad1, m1, den1, h1, acc1);
  k_elu1<<<GRID(NN * H1DIM), B, 0, stream>>>(acc1, b1);

  // ---- layer 2 ----
  k_gemm2<<<strip_blocks, B, 0, stream>>>(acc1, W2, h2);
  k_alpha2<<<GRID(NN), B, 0, stream>>>(h2, a2_src, a2_dst, as2, ad2, m2, den2);
  k_outinit<<<GRID(NN * NCLASS), B, 0, stream>>>(out, b2);
  k_edge_max2<<<GRID(NE), B, 0, stream>>>(ei, as2, ad2, m2);
  k_edge_den2<<<GRID(NE), B, 0, stream>>>(ei, as2, ad2, m2, den2);
  k_edge_scat2<<<GRID(NE * 5), B, 0, stream>>>(ei, as2, ad2, m2, den2, h2, out);
  #undef GRID
}